// MultiGAT_Edge_75952201662545
// MI455X (gfx1250) — compile-verified
//
#include <hip/hip_runtime.h>
#include <hip/hip_bf16.h>
#include <math.h>

#define N_NODES 100000
#define N_EDGES 1600000
#define DIM     128
#define EDIM    8

typedef __attribute__((ext_vector_type(2))) float v2f;
typedef __attribute__((ext_vector_type(8))) float v8f;

// ---------------- float atomic max (correct with -inf init) ----------------
__device__ __forceinline__ void atomic_max_f32(float* addr, float v) {
  if (v >= 0.0f) atomicMax((int*)addr, __float_as_int(v));
  else           atomicMin((unsigned int*)addr, (unsigned int)__float_as_int(v));
}

// ---------------- fold edge_proj + lin_edge into Mt[8][128], b2[128] -------
// ee = edge_attr @ Mt + b2   where Mt[j][c] = sum_i we[c,i]*ep_w[i,j]
__global__ void prep_kernel(const float* __restrict__ we0, const float* __restrict__ epw0,
                            const float* __restrict__ epb0,
                            const float* __restrict__ we1, const float* __restrict__ epw1,
                            const float* __restrict__ epb1,
                            float* __restrict__ Mt0, float* __restrict__ b20,
                            float* __restrict__ Mt1, float* __restrict__ b21) {
  int t = threadIdx.x;                       // 256 threads: 0-127 layer0, 128-255 layer1
  const float* we  = (t < DIM) ? we0  : we1;
  const float* epw = (t < DIM) ? epw0 : epw1;
  const float* epb = (t < DIM) ? epb0 : epb1;
  float* Mt = (t < DIM) ? Mt0 : Mt1;
  float* b2 = (t < DIM) ? b20 : b21;
  int c = t & (DIM - 1);
  float acc[EDIM];
  float accb = 0.0f;
#pragma unroll
  for (int j = 0; j < EDIM; ++j) acc[j] = 0.0f;
  for (int i = 0; i < DIM; ++i) {
    float w = we[c * DIM + i];
    accb += w * epb[i];
#pragma unroll
    for (int j = 0; j < EDIM; ++j) acc[j] += w * epw[i * EDIM + j];
  }
#pragma unroll
  for (int j = 0; j < EDIM; ++j) Mt[j * DIM + c] = acc[j];
  b2[c] = accb;
}

// ---------------- per-layer state init ------------------------------------
__global__ void init_kernel(float* __restrict__ agg, float* __restrict__ m,
                            float* __restrict__ s) {
  int t = blockIdx.x * 256 + threadIdx.x;    // grid covers N*DIM exactly
  agg[t] = 0.0f;
  if (t < N_NODES * 2) { m[t] = -INFINITY; s[t] = 0.0f; }
}

// ---------------- node transforms: xl = x@wl.T, xr = x@wr.T via f32 WMMA ---
// block = 256 thr (8 waves) handles one 16-row tile; wave w owns 16 cols,
// accumulating both wl and wr products (shared A fragment).
__global__ void __launch_bounds__(256) node_wmma_kernel(
    const float* __restrict__ xin, const float* __restrict__ wl,
    const float* __restrict__ wr, float* __restrict__ xl, float* __restrict__ xr) {
  __shared__ __align__(16) float sx[16][132];   // 132-pad -> conflict-free b64 reads
  int t = threadIdx.x;
  int rowbase = blockIdx.x * 16;
  {                                             // stage 16x128 f32 tile (coalesced)
    int r = t >> 4;
    int c = (t & 15) * 8;
    const float4* src = (const float4*)(xin + (size_t)(rowbase + r) * DIM + c);
    float4 a = src[0], b = src[1];
    *(float4*)&sx[r][c]     = a;
    *(float4*)&sx[r][c + 4] = b;
  }
  __syncthreads();

  int lane = t & 31;
  int w    = t >> 5;
  int cw   = w * 16;
  int lh   = lane & 15;
  int hi   = lane >> 4;

  v8f accL = {0.f,0.f,0.f,0.f,0.f,0.f,0.f,0.f};
  v8f accR = {0.f,0.f,0.f,0.f,0.f,0.f,0.f,0.f};
  const float* wlrow = wl + (size_t)(cw + lh) * DIM;   // B[K][N] = wl[cw+N][k+K]
  const float* wrrow = wr + (size_t)(cw + lh) * DIM;

#pragma unroll 4
  for (int k = 0; k < DIM; k += 4) {
    int kk = k + 2 * hi;                               // hi lanes carry K=2,3
    float2 av  = *(const float2*)&sx[lh][kk];          // A: lane lh row, K pair
    float2 blv = *(const float2*)(wlrow + kk);
    float2 brv = *(const float2*)(wrrow + kk);
    v2f a;  a.x = av.x;  a.y = av.y;
    v2f bl; bl.x = blv.x; bl.y = blv.y;
    v2f br; br.x = brv.x; br.y = brv.y;
    accL = __builtin_amdgcn_wmma_f32_16x16x4_f32(false, a, false, bl, (short)0, accL, false, false);
    accR = __builtin_amdgcn_wmma_f32_16x16x4_f32(false, a, false, br, (short)0, accR, false, false);
  }
#pragma unroll
  for (int j = 0; j < 8; ++j) {                        // D: VGPR j = rows j / j+8
    int grow = rowbase + j + 8 * hi;
    xl[(size_t)grow * DIM + cw + lh] = accL[j];
    xr[(size_t)grow * DIM + cw + lh] = accR[j];
  }
}

// ---------------- edge pass 1: alpha + segment max -------------------------
// one wave per edge; lane = 4 channels; lanes 0-15 head0, 16-31 head1
__global__ void __launch_bounds__(256) edge_alpha_kernel(
    const int* __restrict__ ei, const float* __restrict__ eattr,
    const float* __restrict__ Mt, const float* __restrict__ b2,
    const float* __restrict__ att, const float* __restrict__ xl,
    const float* __restrict__ xr, float* __restrict__ ew, float* __restrict__ m) {
  __shared__ __align__(16) float sMt[EDIM * DIM];
  __shared__ __align__(16) float sb2[DIM];
  __shared__ __align__(16) float satt[DIM];
  int t = threadIdx.x;
  for (int i = t; i < EDIM * DIM; i += 256) sMt[i] = Mt[i];
  if (t < DIM) { sb2[t] = b2[t]; satt[t] = att[t]; }
  __syncthreads();

  int lane = t & 31;
  size_t e = (size_t)blockIdx.x * 8 + (t >> 5);
  int src = ei[e];
  int dst = ei[N_EDGES + e];

  float ea[EDIM];
  {
    const float4* p = (const float4*)(eattr + e * EDIM);   // 32B, broadcast
    float4 a0 = p[0], a1 = p[1];
    ea[0]=a0.x; ea[1]=a0.y; ea[2]=a0.z; ea[3]=a0.w;
    ea[4]=a1.x; ea[5]=a1.y; ea[6]=a1.z; ea[7]=a1.w;
  }
  int c0 = lane * 4;
  float4 v = *(const float4*)&sb2[c0];                     // ee on the fly
#pragma unroll
  for (int j = 0; j < EDIM; ++j) {
    float4 mj = *(const float4*)&sMt[j * DIM + c0];
    v.x += ea[j] * mj.x; v.y += ea[j] * mj.y;
    v.z += ea[j] * mj.z; v.w += ea[j] * mj.w;
  }
  float4 xlv = *(const float4*)(xl + (size_t)src * DIM + c0);
  float4 xrv = *(const float4*)(xr + (size_t)dst * DIM + c0);
  v.x += xlv.x + xrv.x; v.y += xlv.y + xrv.y;
  v.z += xlv.z + xrv.z; v.w += xlv.w + xrv.w;
  v.x = v.x > 0.f ? v.x : 0.2f * v.x;                      // leaky_relu
  v.y = v.y > 0.f ? v.y : 0.2f * v.y;
  v.z = v.z > 0.f ? v.z : 0.2f * v.z;
  v.w = v.w > 0.f ? v.w : 0.2f * v.w;
  float4 at = *(const float4*)&satt[c0];
  float p = v.x*at.x + v.y*at.y + v.z*at.z + v.w*at.w;
  p += __shfl_xor(p, 8);                                   // reduce 16-lane half
  p += __shfl_xor(p, 4);
  p += __shfl_xor(p, 2);
  p += __shfl_xor(p, 1);
  if ((lane & 15) == 0) {
    int h = lane >> 4;
    ew[e * 2 + h] = p;
    atomic_max_f32(&m[(size_t)dst * 2 + h], p);
  }
}

// ---------------- edge pass 2: exp + segment sum ---------------------------
__global__ void edge_exp_kernel(const int* __restrict__ ei, float* __restrict__ ew,
                                const float* __restrict__ m, float* __restrict__ s) {
  int t = blockIdx.x * 256 + threadIdx.x;    // covers 2E exactly
  size_t e = (size_t)t >> 1;
  int h = t & 1;
  int dst = ei[N_EDGES + e];
  float ex = __expf(ew[t] - m[(size_t)dst * 2 + h]);
  ew[t] = ex;
  atomicAdd(&s[(size_t)dst * 2 + h], ex);
}

// ---------------- edge pass 3: weighted scatter-add ------------------------
__global__ void __launch_bounds__(256) edge_agg_kernel(
    const int* __restrict__ ei, const float* __restrict__ xl,
    const float* __restrict__ ew, const float* __restrict__ s,
    float* __restrict__ agg) {
  int t = threadIdx.x;
  int lane = t & 31;
  size_t e = (size_t)blockIdx.x * 8 + (t >> 5);
  int src = ei[e];
  int dst = ei[N_EDGES + e];
  int h  = lane >> 4;
  int c0 = lane * 4;
  float aE = ew[e * 2 + h];
  float sd = s[(size_t)dst * 2 + h];
  float a = aE / (sd + 1e-16f);
  float4 xv = *(const float4*)(xl + (size_t)src * DIM + c0);
  float* ag = agg + (size_t)dst * DIM + c0;
  atomicAdd(ag + 0, xv.x * a);
  atomicAdd(ag + 1, xv.y * a);
  atomicAdd(ag + 2, xv.z * a);
  atomicAdd(ag + 3, xv.w * a);
}

// ---------------- bias add (in place) --------------------------------------
__global__ void bias_kernel(float* __restrict__ h, const float* __restrict__ bias) {
  int t = blockIdx.x * 256 + threadIdx.x;    // covers N*DIM exactly
  h[t] += bias[t & (DIM - 1)];
}

// ===========================================================================
extern "C" void kernel_launch(void* const* d_in, const int* in_sizes, int n_in,
                              void* d_out, int out_size, void* d_ws, size_t ws_size,
                              hipStream_t stream) {
  (void)n_in; (void)out_size; (void)ws_size;

  const float *X, *EA;
  const int* EI;
  const float *EPW[2], *EPB[2], *WL[2], *WR[2], *WE[2], *ATT[2], *BIAS[2];

  if (in_sizes[2] == 128) {
    // sorted pytree flatten: edge_attr, edge_index, [att,bias,ep_b,ep_w,we,wl,wr]x2, x
    EA = (const float*)d_in[0];
    EI = (const int*)d_in[1];
    for (int l = 0; l < 2; ++l) {
      int b = 2 + l * 7;
      ATT[l]  = (const float*)d_in[b + 0];
      BIAS[l] = (const float*)d_in[b + 1];
      EPB[l]  = (const float*)d_in[b + 2];
      EPW[l]  = (const float*)d_in[b + 3];
      WE[l]   = (const float*)d_in[b + 4];
      WL[l]   = (const float*)d_in[b + 5];
      WR[l]   = (const float*)d_in[b + 6];
    }
    X = (const float*)d_in[16];
  } else {
    // dict insertion order: x, edge_index, edge_attr, [ep_w,ep_b,wl,wr,we,att,bias]x2
    X  = (const float*)d_in[0];
    EI = (const int*)d_in[1];
    EA = (const float*)d_in[2];
    for (int l = 0; l < 2; ++l) {
      int b = 3 + l * 7;
      EPW[l]  = (const float*)d_in[b + 0];
      EPB[l]  = (const float*)d_in[b + 1];
      WL[l]   = (const float*)d_in[b + 2];
      WR[l]   = (const float*)d_in[b + 3];
      WE[l]   = (const float*)d_in[b + 4];
      ATT[l]  = (const float*)d_in[b + 5];
      BIAS[l] = (const float*)d_in[b + 6];
    }
  }

  // workspace layout (f32): xl | xr | h | ew | m | s | Mt0 b20 Mt1 b21
  float* ws = (float*)d_ws;
  size_t nd = (size_t)N_NODES * DIM;
  float* xl   = ws;
  float* xr   = ws + nd;
  float* hbuf = ws + 2 * nd;
  float* ew   = ws + 3 * nd;
  float* m    = ew + (size_t)N_EDGES * 2;
  float* s    = m + (size_t)N_NODES * 2;
  float* Mt0  = s + (size_t)N_NODES * 2;
  float* b20  = Mt0 + EDIM * DIM;
  float* Mt1  = b20 + DIM;
  float* b21  = Mt1 + EDIM * DIM;

  hipLaunchKernelGGL(prep_kernel, dim3(1), dim3(256), 0, stream,
                     WE[0], EPW[0], EPB[0], WE[1], EPW[1], EPB[1],
                     Mt0, b20, Mt1, b21);

  const float* xin_l[2]  = { X, hbuf };
  float*       agg_l[2]  = { hbuf, (float*)d_out };
  const float* Mt_l[2]   = { Mt0, Mt1 };
  const float* b2_l[2]   = { b20, b21 };

  const int blkN   = (int)(nd / 256);          // 50000
  const int blkRow = N_NODES / 16;             // 6250
  const int blkE   = N_EDGES / 8;              // 200000
  const int blk2E  = (2 * N_EDGES) / 256;      // 12500

  for (int l = 0; l < 2; ++l) {
    hipLaunchKernelGGL(init_kernel, dim3(blkN), dim3(256), 0, stream,
                       agg_l[l], m, s);
    hipLaunchKernelGGL(node_wmma_kernel, dim3(blkRow), dim3(256), 0, stream,
                       xin_l[l], WL[l], WR[l], xl, xr);
    hipLaunchKernelGGL(edge_alpha_kernel, dim3(blkE), dim3(256), 0, stream,
                       EI, EA, Mt_l[l], b2_l[l], ATT[l], xl, xr, ew, m);
    hipLaunchKernelGGL(edge_exp_kernel, dim3(blk2E), dim3(256), 0, stream,
                       EI, ew, m, s);
    hipLaunchKernelGGL(edge_agg_kernel, dim3(blkE), dim3(256), 0, stream,
                       EI, xl, ew, s, agg_l[l]);
    hipLaunchKernelGGL(bias_kernel, dim3(blkN), dim3(256), 0, stream,
                       agg_l[l], BIAS[l]);
  }
}